// Self_Attn_40656160424177
// MI455X (gfx1250) — compile-verified
//
#include <hip/hip_runtime.h>

#define BATCH 8
#define CCH   128            // channels
#define NPIX  3136           // 56*56
#define NT    196            // NPIX / 16
#define NT2   98             // NT / 2
#define CT    8              // CCH  / 16

typedef float v2f __attribute__((ext_vector_type(2)));
typedef float v8f __attribute__((ext_vector_type(8)));

__device__ __forceinline__ v8f wmma_f32(const v2f& a, const v2f& b, const v8f& c) {
  return __builtin_amdgcn_wmma_f32_16x16x4_f32(false, a, false, b,
                                               (short)0, c, false, false);
}

// Store one 16x16 f32 C/D tile: lane l, vgpr v -> row = v + 8*(l>=16), col = l&15
__device__ __forceinline__ void store_tile(float* __restrict__ E, const v8f& acc,
                                           int ti, int tj, int lane) {
  const int rbase = ti * 16 + ((lane >> 4) << 3);
  const int col   = tj * 16 + (lane & 15);
#pragma unroll
  for (int v = 0; v < 8; ++v)
    E[(size_t)(rbase + v) * NPIX + col] = acc[v];
}

// ---------------------------------------------------------------------------
// Kernel 1: energy[i,j] = sum_c x[b,c,i] * x[b,c,j]   (Gram, K=C=128)
// One wave per 32x32 output block (2x2 tiles, 4 accumulators).
// Per k-step: 8 b32 loads feed 4 independent WMMAs (2 loads/WMMA).
// ---------------------------------------------------------------------------
__global__ __launch_bounds__(128) void energy_kernel(const float* __restrict__ x,
                                                     float* __restrict__ E, int b) {
  const int wave = blockIdx.x * 4 + (threadIdx.x >> 5);    // exact: NT2*NT2 waves
  const int ti0  = (wave / NT2) * 2;
  const int tj0  = (wave % NT2) * 2;
  const int lane = threadIdx.x & 31;
  const int mm   = lane & 15;
  const int kh   = (lane >> 4) << 1;                       // 0 or 2

  const float* xb  = x + (size_t)b * CCH * NPIX;
  const float* pa0 = xb + ti0 * 16 + mm;                   // + (c+k)*NPIX
  const float* pa1 = pa0 + 16;
  const float* pb0 = xb + tj0 * 16 + mm;
  const float* pb1 = pb0 + 16;

  v8f acc00 = {}, acc01 = {}, acc10 = {}, acc11 = {};
#pragma unroll 4
  for (int c = 0; c < CCH; c += 4) {
    const size_t r0 = (size_t)(c + kh) * NPIX;
    const size_t r1 = (size_t)(c + kh + 1) * NPIX;
    v2f a0, a1, b0, b1;
    a0.x = pa0[r0];  a0.y = pa0[r1];
    a1.x = pa1[r0];  a1.y = pa1[r1];
    b0.x = pb0[r0];  b0.y = pb0[r1];
    b1.x = pb1[r0];  b1.y = pb1[r1];
    acc00 = wmma_f32(a0, b0, acc00);
    acc01 = wmma_f32(a0, b1, acc01);
    acc10 = wmma_f32(a1, b0, acc10);
    acc11 = wmma_f32(a1, b1, acc11);
  }

  store_tile(E, acc00, ti0,     tj0,     lane);
  store_tile(E, acc01, ti0,     tj0 + 1, lane);
  store_tile(E, acc10, ti0 + 1, tj0,     lane);
  store_tile(E, acc11, ti0 + 1, tj0 + 1, lane);
}

// ---------------------------------------------------------------------------
// Kernel 2: row softmax with diagonal zeroed BEFORE softmax (mult. mask).
// One 256-thread block (8 wave32s) per row; in-place on E.
// ---------------------------------------------------------------------------
__device__ __forceinline__ float wave_max(float v) {
#pragma unroll
  for (int o = 16; o; o >>= 1) v = fmaxf(v, __shfl_xor(v, o, 32));
  return v;
}
__device__ __forceinline__ float wave_sum(float v) {
#pragma unroll
  for (int o = 16; o; o >>= 1) v += __shfl_xor(v, o, 32);
  return v;
}

__global__ __launch_bounds__(256) void softmax_kernel(float* __restrict__ E) {
  const int i   = blockIdx.x;
  const int tid = threadIdx.x;
  float* row = E + (size_t)i * NPIX;
  __shared__ float red[8];

  float m = -3.402823466e38f;
  for (int j = tid; j < NPIX; j += 256) {
    float v = (j == i) ? 0.0f : row[j];
    m = fmaxf(m, v);
  }
  m = wave_max(m);
  if ((tid & 31) == 0) red[tid >> 5] = m;
  __syncthreads();
  float gm = red[0];
#pragma unroll
  for (int k = 1; k < 8; ++k) gm = fmaxf(gm, red[k]);
  __syncthreads();

  float s = 0.0f;
  for (int j = tid; j < NPIX; j += 256) {
    float v = (j == i) ? 0.0f : row[j];
    float e = __expf(v - gm);
    row[j] = e;
    s += e;
  }
  s = wave_sum(s);
  if ((tid & 31) == 0) red[tid >> 5] = s;
  __syncthreads();
  float gs = 0.0f;
#pragma unroll
  for (int k = 0; k < 8; ++k) gs += red[k];
  const float inv = 1.0f / gs;

  for (int j = tid; j < NPIX; j += 256) row[j] *= inv;
}

// ---------------------------------------------------------------------------
// Kernel 3: out[c,j] = sum_i V[c,i] * A[i,j]  (M=128, N=3136, K=3136)
// One wave per 32x32 output block (2 ct x 2 tj, 4 accumulators).
// Per k-step: 2 b64 (V, k-pair contiguous) + 4 b32 (attention) feed 4 WMMAs.
// Fused epilogue: relu, y = g*out + x, out copy, x passthrough.
// ---------------------------------------------------------------------------
__global__ __launch_bounds__(128) void out_kernel(const float* __restrict__ x,
                                                  const float* __restrict__ A,
                                                  const float* __restrict__ gammap,
                                                  float* __restrict__ y,
                                                  float* __restrict__ outb,
                                                  float* __restrict__ xc, int b) {
  const int wave = blockIdx.x * 4 + (threadIdx.x >> 5);    // exact: 4*NT2 waves
  const int ct0  = (wave / NT2) * 2;                       // 0,2,4,6
  const int tj0  = (wave % NT2) * 2;
  const int lane = threadIdx.x & 31;
  const int mm   = lane & 15;
  const int kh   = (lane >> 4) << 1;

  const float* xb  = x + (size_t)b * CCH * NPIX;
  const float* pv0 = xb + (size_t)(ct0 * 16 + mm) * NPIX + kh;       // + i0
  const float* pv1 = pv0 + (size_t)16 * NPIX;
  const float* pa0 = A + tj0 * 16 + mm;                              // + (i0+k)*NPIX
  const float* pa1 = pa0 + 16;

  v8f acc00 = {}, acc01 = {}, acc10 = {}, acc11 = {};
#pragma unroll 2
  for (int i0 = 0; i0 < NPIX; i0 += 4) {
    v2f a0 = *(const v2f*)(pv0 + i0);                      // V[ct0 tile , i-pair]
    v2f a1 = *(const v2f*)(pv1 + i0);                      // V[ct0+1 tile, i-pair]
    const size_t r0 = (size_t)(i0 + kh) * NPIX;
    const size_t r1 = (size_t)(i0 + kh + 1) * NPIX;
    v2f b0, b1;
    b0.x = pa0[r0];  b0.y = pa0[r1];
    b1.x = pa1[r0];  b1.y = pa1[r1];
    acc00 = wmma_f32(a0, b0, acc00);
    acc01 = wmma_f32(a0, b1, acc01);
    acc10 = wmma_f32(a1, b0, acc10);
    acc11 = wmma_f32(a1, b1, acc11);
  }

  const float g = *gammap;
  const v8f* accs[4] = { &acc00, &acc01, &acc10, &acc11 };
#pragma unroll
  for (int t = 0; t < 4; ++t) {
    const int ct = ct0 + (t >> 1);
    const int tj = tj0 + (t & 1);
    const int cbase = ct * 16 + ((lane >> 4) << 3);
    const int j     = tj * 16 + mm;
#pragma unroll
    for (int v = 0; v < 8; ++v) {
      const size_t loc = (size_t)(cbase + v) * NPIX + j;
      const size_t idx = (size_t)b * CCH * NPIX + loc;
      const float  o   = fmaxf((*accs[t])[v], 0.0f);       // relu
      const float  xv  = xb[loc];
      y[idx]    = g * o + xv;                              // y = gamma*out + x
      outb[idx] = o;                                       // out
      xc[idx]   = xv;                                      // x passthrough
    }
  }
}

__global__ void write_gamma_kernel(const float* __restrict__ g, float* __restrict__ dst) {
  if (threadIdx.x == 0) *dst = *g;
}

// ---------------------------------------------------------------------------
// Host launch: per-batch pipeline keeps the 39.3MB energy matrix L2-resident.
// ---------------------------------------------------------------------------
extern "C" void kernel_launch(void* const* d_in, const int* in_sizes, int n_in,
                              void* d_out, int out_size, void* d_ws, size_t ws_size,
                              hipStream_t stream) {
  const float* x     = (const float*)d_in[0];
  const float* gamma = (const float*)d_in[1];
  float* out = (float*)d_out;
  float* E   = (float*)d_ws;                               // N*N fp32 = 39.3 MB

  const size_t S = (size_t)BATCH * CCH * NPIX;             // 3,211,264
  float* y_ptr   = out;
  float* out_ptr = out + S;
  float* x_ptr   = out + 2 * S;
  float* g_ptr   = out + 3 * S;

  write_gamma_kernel<<<1, 1, 0, stream>>>(gamma, g_ptr);

  for (int b = 0; b < BATCH; ++b) {
    energy_kernel<<<(NT2 * NT2) / 4, 128, 0, stream>>>(x, E, b);      // 2401 blocks
    softmax_kernel<<<NPIX, 256, 0, stream>>>(E);                      // 3136 rows
    out_kernel<<<(4 * NT2) / 4, 128, 0, stream>>>(x, E, gamma,
                                                  y_ptr, out_ptr, x_ptr, b); // 98 blocks
  }
}